// GCN2_41154376630906
// MI455X (gfx1250) — compile-verified
//
#include <hip/hip_runtime.h>
#include <hip/hip_bf16.h>

typedef __attribute__((ext_vector_type(2))) float v2f;
typedef __attribute__((ext_vector_type(8))) float v8f;

#define C_IN  128
#define C_HID 64
#define C_OUT 2
#define WPAD  132   // padded K-stride for transposed W1 in LDS (bank-conflict free)

__device__ __forceinline__ void atomic_add_f32(float* p, float v) {
    // lowers to global_atomic_add_f32 (relaxed, device scope)
    __hip_atomic_fetch_add(p, v, __ATOMIC_RELAXED, __HIP_MEMORY_SCOPE_AGENT);
}

// ---- degree: deg[i] = 1 (self loop) + in-degree ----------------------------
__global__ void k_init_deg(float* __restrict__ deg, int n) {
    int i = blockIdx.x * blockDim.x + threadIdx.x;
    if (i < n) deg[i] = 1.0f;
}

__global__ void k_count_deg(const int* __restrict__ ei, float* __restrict__ deg, int E) {
    int e = blockIdx.x * blockDim.x + threadIdx.x;
    if (e < E) atomic_add_f32(&deg[ei[(size_t)E + e]], 1.0f);  // targets = col
}

__global__ void k_dinv(float* __restrict__ deg, int n) {
    int i = blockIdx.x * blockDim.x + threadIdx.x;
    if (i < n) deg[i] = __frsqrt_rn(deg[i]);   // deg >= 1 always
}

// ---- GEMM1: h1[N,64] = x[N,128] @ W1[128,64] via V_WMMA_F32_16X16X4_F32 ----
// One wave per PAIR of 16-node tiles (32 rows): each ds_load_b64 of a B
// operand feeds two WMMAs. W1 staged transposed+padded so B pairs are
// contiguous float2 (single ds_load_b64, no repacking movs).
__global__ __launch_bounds__(256)
void k_gemm1_wmma(const float* __restrict__ x, const float* __restrict__ W1,
                  float* __restrict__ h1, int numPairs) {
    __shared__ float Wt[C_HID * WPAD];                  // ~33 KB of 320 KB WGP LDS
    for (int i = threadIdx.x; i < C_HID * C_IN; i += 256) {
        int nn = i >> 7;        // output channel 0..63
        int kk = i & 127;       // k 0..127
        Wt[nn * WPAD + kk] = W1[kk * C_HID + nn];
    }
    __syncthreads();

    int wave = threadIdx.x >> 5;
    int lane = threadIdx.x & 31;
    int pair = blockIdx.x * 8 + wave;
    if (pair >= numPairs) pair = numPairs - 1;          // benign duplicate

    int m     = lane & 15;      // A-matrix row (M) held by this lane
    int khalf = lane >> 4;      // K offset select per ISA VGPR layout
    int ncol  = lane & 15;      // B/D column within 16-wide subtile

    const float* xrow0 = x + ((size_t)pair * 32 + m) * C_IN + 2 * khalf;
    const float* xrow1 = xrow0 + (size_t)16 * C_IN;

    // B 4x16 f32: vgpr v, lane L -> N = L&15 (+16*ntile), K = k0 + v + 2*(L>>4)
    // In Wt layout that is a contiguous float2 at [n*WPAD + k0 + 2*khalf].
    const v2f* bp0 = (const v2f*)&Wt[(ncol     ) * WPAD + 2 * khalf];
    const v2f* bp1 = (const v2f*)&Wt[(ncol + 16) * WPAD + 2 * khalf];
    const v2f* bp2 = (const v2f*)&Wt[(ncol + 32) * WPAD + 2 * khalf];
    const v2f* bp3 = (const v2f*)&Wt[(ncol + 48) * WPAD + 2 * khalf];

    v8f a00 = {}, a01 = {}, a02 = {}, a03 = {};   // tile 0 accumulators
    v8f a10 = {}, a11 = {}, a12 = {}, a13 = {};   // tile 1 accumulators
    #pragma unroll 4
    for (int k0 = 0; k0 < C_IN; k0 += 4) {
        // A 16x4 f32: vgpr v, lane L -> M = L&15, K = k0 + v + 2*(L>>4)
        v2f av0 = *(const v2f*)(xrow0 + k0);
        v2f av1 = *(const v2f*)(xrow1 + k0);
        v2f b0 = bp0[k0 >> 1];
        v2f b1 = bp1[k0 >> 1];
        v2f b2 = bp2[k0 >> 1];
        v2f b3 = bp3[k0 >> 1];
        a00 = __builtin_amdgcn_wmma_f32_16x16x4_f32(false, av0, false, b0, (short)0, a00, false, false);
        a10 = __builtin_amdgcn_wmma_f32_16x16x4_f32(false, av1, false, b0, (short)0, a10, false, false);
        a01 = __builtin_amdgcn_wmma_f32_16x16x4_f32(false, av0, false, b1, (short)0, a01, false, false);
        a11 = __builtin_amdgcn_wmma_f32_16x16x4_f32(false, av1, false, b1, (short)0, a11, false, false);
        a02 = __builtin_amdgcn_wmma_f32_16x16x4_f32(false, av0, false, b2, (short)0, a02, false, false);
        a12 = __builtin_amdgcn_wmma_f32_16x16x4_f32(false, av1, false, b2, (short)0, a12, false, false);
        a03 = __builtin_amdgcn_wmma_f32_16x16x4_f32(false, av0, false, b3, (short)0, a03, false, false);
        a13 = __builtin_amdgcn_wmma_f32_16x16x4_f32(false, av1, false, b3, (short)0, a13, false, false);
    }

    // D 16x16 f32: vgpr r, lanes 0-15 -> M=r, lanes 16-31 -> M=r+8; N = L&15
    float* h0 = h1 + (size_t)pair * 32 * C_HID;
    float* h16 = h0 + (size_t)16 * C_HID;
    #pragma unroll
    for (int r = 0; r < 8; ++r) {
        int mrow = r + 8 * khalf;
        h0[mrow * C_HID + ncol]       = a00[r];
        h0[mrow * C_HID + 16 + ncol]  = a01[r];
        h0[mrow * C_HID + 32 + ncol]  = a02[r];
        h0[mrow * C_HID + 48 + ncol]  = a03[r];
        h16[mrow * C_HID + ncol]      = a10[r];
        h16[mrow * C_HID + 16 + ncol] = a11[r];
        h16[mrow * C_HID + 32 + ncol] = a12[r];
        h16[mrow * C_HID + 48 + ncol] = a13[r];
    }
}

// ---- agg1 init: bias + self-loop term --------------------------------------
__global__ void k_init_agg1(const float* __restrict__ h1, const float* __restrict__ dinv,
                            const float* __restrict__ b1, float* __restrict__ agg1, int total) {
    int idx = blockIdx.x * blockDim.x + threadIdx.x;
    if (idx >= total) return;
    int i = idx >> 6;            // node
    int c = idx & 63;            // channel
    float d = dinv[i];
    agg1[idx] = b1[c] + d * d * h1[idx];
}

// ---- agg1 edges: one wave per edge, 2 coalesced channels per lane ----------
__global__ __launch_bounds__(256)
void k_agg1_edges(const int* __restrict__ ei, const float* __restrict__ dinv,
                  const float* __restrict__ h1, float* __restrict__ agg1, int E) {
    int gw   = (blockIdx.x * blockDim.x + threadIdx.x) >> 5;  // edge id
    int lane = threadIdx.x & 31;
    if (gw >= E) return;
    int row = ei[gw];
    int col = ei[(size_t)E + gw];
    float norm = dinv[row] * dinv[col];
    const float* hr = h1 + (size_t)row * C_HID;
    float* ac = agg1 + (size_t)col * C_HID;
    atomic_add_f32(&ac[lane],      norm * hr[lane]);
    atomic_add_f32(&ac[lane + 32], norm * hr[lane + 32]);
}

// ---- layer 2: h2 = relu(agg1) @ W2; out = b2 + dinv^2 * h2 (self loop) -----
__global__ void k_gemm2(const float* __restrict__ agg1, const float* __restrict__ W2,
                        const float* __restrict__ b2, const float* __restrict__ dinv,
                        float* __restrict__ h2, float* __restrict__ out, int n) {
    int i = blockIdx.x * blockDim.x + threadIdx.x;
    if (i >= n) return;
    const float* a = agg1 + (size_t)i * C_HID;
    float o0 = 0.f, o1 = 0.f;
    #pragma unroll 8
    for (int k = 0; k < C_HID; ++k) {
        float v = a[k];
        v = v > 0.f ? v : 0.f;                    // ReLU fused here
        o0 = fmaf(v, W2[k * 2 + 0], o0);
        o1 = fmaf(v, W2[k * 2 + 1], o1);
    }
    h2[i * 2 + 0] = o0;
    h2[i * 2 + 1] = o1;
    float d2 = dinv[i] * dinv[i];
    out[i * 2 + 0] = b2[0] + d2 * o0;
    out[i * 2 + 1] = b2[1] + d2 * o1;
}

// ---- agg2 edges: one thread per edge, 2 channels ---------------------------
__global__ void k_agg2_edges(const int* __restrict__ ei, const float* __restrict__ dinv,
                             const float* __restrict__ h2, float* __restrict__ out, int E) {
    int e = blockIdx.x * blockDim.x + threadIdx.x;
    if (e >= E) return;
    int row = ei[e];
    int col = ei[(size_t)E + e];
    float norm = dinv[row] * dinv[col];
    atomic_add_f32(&out[col * 2 + 0], norm * h2[row * 2 + 0]);
    atomic_add_f32(&out[col * 2 + 1], norm * h2[row * 2 + 1]);
}

extern "C" void kernel_launch(void* const* d_in, const int* in_sizes, int n_in,
                              void* d_out, int out_size, void* d_ws, size_t ws_size,
                              hipStream_t stream) {
    const float* x  = (const float*)d_in[0];
    const int*   ei = (const int*)d_in[1];      // edge_index [2,E]
    const float* W1 = (const float*)d_in[2];
    const float* b1 = (const float*)d_in[3];
    const float* W2 = (const float*)d_in[4];
    const float* b2 = (const float*)d_in[5];
    float* out = (float*)d_out;

    int n = in_sizes[0] / C_IN;    // 100000
    int E = in_sizes[1] / 2;       // 3200000

    float* ws   = (float*)d_ws;
    float* dinv = ws;                               // n
    float* h1   = dinv + n;                         // n * 64
    float* agg1 = h1 + (size_t)n * C_HID;           // n * 64
    float* h2   = agg1 + (size_t)n * C_HID;         // n * 2

    const int B = 256;
    int numPairs = (n + 31) / 32;  // 32 rows per wave

    k_init_deg  <<<(n + B - 1) / B, B, 0, stream>>>(dinv, n);
    k_count_deg <<<(E + B - 1) / B, B, 0, stream>>>(ei, dinv, E);
    k_dinv      <<<(n + B - 1) / B, B, 0, stream>>>(dinv, n);

    k_gemm1_wmma<<<(numPairs + 7) / 8, B, 0, stream>>>(x, W1, h1, numPairs);

    int tot1 = n * C_HID;
    k_init_agg1 <<<(tot1 + B - 1) / B, B, 0, stream>>>(h1, dinv, b1, agg1, tot1);
    k_agg1_edges<<<((size_t)E * 32 + B - 1) / B, B, 0, stream>>>(ei, dinv, h1, agg1, E);

    k_gemm2     <<<(n + B - 1) / B, B, 0, stream>>>(agg1, W2, b2, dinv, h2, out, n);
    k_agg2_edges<<<(E + B - 1) / B, B, 0, stream>>>(ei, dinv, h2, out, E);
}